// CollapsePreventionLoss_63127429317292
// MI455X (gfx1250) — compile-verified
//
#include <hip/hip_runtime.h>

typedef __attribute__((ext_vector_type(2))) float v2f;
typedef __attribute__((ext_vector_type(8))) float v8f;

#define N_ATOMS 1024
#define TILE    16
#define NT      64          // N_ATOMS / TILE
#define NPAIRS  2080        // NT*(NT+1)/2 upper-triangular tile pairs
#define SPLITS  16
#define MIN_DIST 2.9f

__global__ void collapse_zero_kernel(float* out) {
    if (threadIdx.x == 0) out[0] = 0.0f;
}

__global__ __launch_bounds__(256)
void collapse_loss_kernel(const float* __restrict__ coords, float* __restrict__ out, int B)
{
    __shared__ float4 satoms[N_ATOMS];   // x, y, z, |x|^2

    const int block = blockIdx.x;
    const int b     = block / SPLITS;
    const int split = block % SPLITS;
    const int tid   = threadIdx.x;
    const int lane  = tid & 31;
    const int wave  = tid >> 5;

    // ---- stage batch coords + norms into LDS (float4 -> ds_load_b128) ----
    const float* cb = coords + (size_t)b * (N_ATOMS * 3);
    for (int a = tid; a < N_ATOMS; a += 256) {
        float x = cb[a * 3 + 0];
        float y = cb[a * 3 + 1];
        float z = cb[a * 3 + 2];
        satoms[a] = make_float4(x, y, z, x * x + y * y + z * z);
    }
    __syncthreads();

    const int tiles_per_block = (NPAIRS + SPLITS - 1) / SPLITS;  // 130
    const int t0 = split * tiles_per_block;
    const int t1 = (t0 + tiles_per_block < NPAIRS) ? (t0 + tiles_per_block) : NPAIRS;

    float acc = 0.0f;

    for (int t = t0 + wave; t < t1; t += 8) {
        // ---- map linear tile index -> (r, c) in upper triangle (incl. diagonal) ----
        float tf = (float)t;
        int r = (int)((2.0f * NT + 1.0f -
                       __builtin_amdgcn_sqrtf((2.0f * NT + 1.0f) * (2.0f * NT + 1.0f) - 8.0f * tf)) * 0.5f);
        if (r < 0) r = 0;
        if (r > NT - 1) r = NT - 1;
        while (r * NT - (r * (r - 1)) / 2 > t) --r;
        while ((r + 1) * NT - ((r + 1) * r) / 2 <= t) ++r;
        const int c  = t - (r * NT - (r * (r - 1)) / 2) + r;
        const int i0 = r * TILE;
        const int j0 = c * TILE;

        // ---- load fragments from LDS (branch-free: EXEC stays all-ones) ----
        const float4 fa = satoms[i0 + (lane & 15)];
        const float4 fb = satoms[j0 + (lane & 15)];
        const bool lohalf = (lane < 16);
        v2f afrag, bfrag;
        afrag.x = lohalf ? fa.x : fa.z;
        afrag.y = lohalf ? fa.y : 0.0f;
        bfrag.x = lohalf ? fb.x : fb.z;
        bfrag.y = lohalf ? fb.y : 0.0f;

        // ---- Gram tile: D = A(16x4) * B(4x16), f32 WMMA ----
        v8f dmat = {};
        dmat = __builtin_amdgcn_wmma_f32_16x16x4_f32(
            /*neg_a=*/false, afrag, /*neg_b=*/false, bfrag,
            /*c_mod=*/(short)0, dmat, /*reuse_a=*/false, /*reuse_b=*/false);

        // ---- epilogue: dist^2 = ni + nj - 2*g, hinge ----
        const float nj = fb.w;                 // norm of column atom (same for all 8 rows)
        const int   hi = (lane >> 4) & 1;

        if (r != c) {
            // Off-diagonal tile (97% of tiles): i < j holds for every element,
            // no mask needed. Wave-uniform branch -> s_cbranch, EXEC untouched.
            #pragma unroll
            for (int rr = 0; rr < 8; ++rr) {
                const int   M  = rr + hi * 8;
                const float ni = __shfl(fa.w, M, 32);   // row-norm broadcast (wave32)
                float dsq = ni + nj - 2.0f * dmat[rr];
                dsq = fmaxf(dsq, 0.0f);
                const float dist = __builtin_amdgcn_sqrtf(dsq + 1e-8f);
                const float v = fmaxf(MIN_DIST - dist, 0.0f);
                acc = fmaf(v, v, acc);
            }
        } else {
            // Diagonal tile: apply strict upper-triangle mask per element.
            const int jg = j0 + (lane & 15);
            #pragma unroll
            for (int rr = 0; rr < 8; ++rr) {
                const int   M  = rr + hi * 8;
                const float ni = __shfl(fa.w, M, 32);
                const int   ig = i0 + M;
                float dsq = ni + nj - 2.0f * dmat[rr];
                dsq = fmaxf(dsq, 0.0f);
                const float dist = __builtin_amdgcn_sqrtf(dsq + 1e-8f);
                float v = fmaxf(MIN_DIST - dist, 0.0f);
                v = (ig < jg) ? v : 0.0f;
                acc = fmaf(v, v, acc);
            }
        }
    }

    // ---- wave32 reduction, one atomic per wave ----
    for (int off = 16; off > 0; off >>= 1)
        acc += __shfl_xor(acc, off, 32);
    if (lane == 0)
        atomicAdd(out, acc * (1.0f / (float)B));
}

extern "C" void kernel_launch(void* const* d_in, const int* in_sizes, int n_in,
                              void* d_out, int out_size, void* d_ws, size_t ws_size,
                              hipStream_t stream) {
    const float* coords = (const float*)d_in[0];
    float* out = (float*)d_out;

    const int total = in_sizes[0];          // B * 3072
    const int B = total / (N_ATOMS * 3);    // = 64 for the reference shapes

    collapse_zero_kernel<<<1, 32, 0, stream>>>(out);
    collapse_loss_kernel<<<B * SPLITS, 256, 0, stream>>>(coords, out, B);
}